// GraphQNNHybrid_65481071398469
// MI455X (gfx1250) — compile-verified
//
#include <hip/hip_runtime.h>
#include <math.h>

#define N_NODES 50000
#define N_EDGES 400000
#define EN (N_EDGES + N_NODES)   // edges + self loops
#define H 4
#define C 128
#define HC 512                   // H*C
#define SLOPE 0.2f

typedef __attribute__((ext_vector_type(2))) float v2f;
typedef __attribute__((ext_vector_type(8))) float v8f;

// ---------------------------------------------------------------------------
// float atomic-max via sign-split integer atomics (works with -inf init)
// ---------------------------------------------------------------------------
__device__ __forceinline__ void atomicMaxF(float* addr, float val) {
    if (val >= 0.0f)
        atomicMax((int*)addr, __float_as_int(val));
    else
        atomicMin((unsigned int*)addr, __float_as_uint(val));
}

// ---------------------------------------------------------------------------
// init: emax = -inf, denom = 0, agg = 0
// ---------------------------------------------------------------------------
__global__ void init_kernel(float* __restrict__ emax, float* __restrict__ denom,
                            float* __restrict__ agg) {
    const size_t stride = (size_t)gridDim.x * blockDim.x;
    size_t i = (size_t)blockIdx.x * blockDim.x + threadIdx.x;
    if (i < (size_t)N_NODES * H) {
        emax[i]  = -INFINITY;
        denom[i] = 0.0f;
    }
    const size_t total = (size_t)N_NODES * HC;
    for (size_t j = i; j < total; j += stride) agg[j] = 0.0f;
}

// ---------------------------------------------------------------------------
// WMMA fp32 GEMM: D[M,Nout] = act(A[M,K] @ Wm[Nout,K]^T + bias (+ resid))
// MODE 0: plain        MODE 1: +bias +resid, ReLU      MODE 2: +bias, tanh
// Block: 128 threads = 4 waves. Each wave owns TN accumulators covering a
// 16 x (TN*16) strip of D; the 4 waves tile N within the block.
// Per k-step: 1 LDS A-fragment load + TN global B loads + TN WMMAs
//   -> (1+TN)/TN loads per WMMA, TN independent acc chains for the XDL pipe.
// Grid: (M/16, Nout/(64*TN)). M, K, Nout multiples of 16; no masking, so
// EXEC is all-ones across every WMMA as the ISA requires.
// ---------------------------------------------------------------------------
template <int MODE, int TN>
__global__ __launch_bounds__(128)
void gemm_wmma(const float* __restrict__ A, const float* __restrict__ Wm,
               const float* __restrict__ bias, const float* __restrict__ resid,
               float* __restrict__ D, int K, int Nout) {
    __shared__ float lds_a[16 * 512];   // A tile: 16 rows x K (K <= 512)

    const int tid  = threadIdx.x;
    const int wave = tid >> 5;
    const int lane = tid & 31;
    const int m0   = blockIdx.x * 16;
    const int n0   = (blockIdx.y * 4 + wave) * (TN * 16);

    // Stage A tile: rows [m0, m0+16) x full K width == contiguous 16*K floats.
    {
        const float4* srcA = (const float4*)(A + (size_t)m0 * K);
        float4* dstA = (float4*)lds_a;
        const int n4 = 4 * K;                      // 16*K/4 float4 elements
        for (int i = tid; i < n4; i += 128) dstA[i] = srcA[i];
    }
    __syncthreads();

    // Fragment addressing per CDNA5 ISA 7.12.2 (32-bit 16x4 A / 4x16 B):
    //   VGPR v, lanes 0-15:  K = v       ; lanes 16-31: K = v + 2
    const int lhalf = lane >> 4;     // 0 | 1
    const int lmod  = lane & 15;
    const float* arow = lds_a + (size_t)lmod * K + 2 * lhalf;
    const float* brow = Wm + (size_t)(n0 + lmod) * K + 2 * lhalf;
    const int bstride = 16 * K;      // float offset between adjacent N tiles

    v8f acc[TN];
#pragma unroll
    for (int t = 0; t < TN; ++t) acc[t] = (v8f){0.f,0.f,0.f,0.f,0.f,0.f,0.f,0.f};

#pragma unroll 4
    for (int k = 0; k < K; k += 4) {
        const v2f a = *(const v2f*)(arow + k);              // ds_load_b64
#pragma unroll
        for (int t = 0; t < TN; ++t) {
            const v2f b = *(const v2f*)(brow + t * bstride + k);  // global_load_b64
            acc[t] = __builtin_amdgcn_wmma_f32_16x16x4_f32(
                /*neg_a=*/false, a, /*neg_b=*/false, b,
                /*c_mod=*/(short)0, acc[t], /*reuse_a=*/false, /*reuse_b=*/false);
        }
    }

    // D layout: VGPR v -> row = v + 8*lhalf, col = lmod (within tile)
#pragma unroll
    for (int t = 0; t < TN; ++t) {
#pragma unroll
        for (int v = 0; v < 8; ++v) {
            const int row = m0 + v + 8 * lhalf;
            const int col = n0 + t * 16 + lmod;
            float val = acc[t][v];
            if (MODE != 0) val += bias[col];
            if (MODE == 1) {
                val += resid[(size_t)row * Nout + col];
                val = fmaxf(val, 0.0f);
            }
            if (MODE == 2) val = tanhf(val);
            D[(size_t)row * Nout + col] = val;
        }
    }
}

// ---------------------------------------------------------------------------
// per-(node,head) attention scores: a_s = <h, att_src>, a_d = <h, att_dst>
// ---------------------------------------------------------------------------
__global__ void att_score_kernel(const float* __restrict__ h,
                                 const float* __restrict__ att_src,
                                 const float* __restrict__ att_dst,
                                 float* __restrict__ a_s, float* __restrict__ a_d) {
    const int id = blockIdx.x * blockDim.x + threadIdx.x;
    if (id >= N_NODES * H) return;
    const int i  = id >> 2;
    const int hh = id & 3;
    const float* hr = h + (size_t)i * HC + hh * C;
    const float* as = att_src + hh * C;
    const float* ad = att_dst + hh * C;
    float s = 0.0f, d = 0.0f;
    for (int c = 0; c < C; ++c) {
        const float hv = hr[c];
        s += hv * as[c];
        d += hv * ad[c];
    }
    a_s[id] = s;
    a_d[id] = d;
}

__device__ __forceinline__ void edge_endpoints(const int* ei, int e, int& src, int& dst) {
    if (e < N_EDGES) { src = ei[e]; dst = ei[N_EDGES + e]; }
    else             { src = dst = e - N_EDGES; }     // self loop
}

// pass 1: e = leaky_relu(a_s[src] + a_d[dst]); segment max over dst
__global__ void edge_pass1(const int* __restrict__ ei, const float* __restrict__ a_s,
                           const float* __restrict__ a_d, float* __restrict__ ebuf,
                           float* __restrict__ emax) {
    const int id = blockIdx.x * blockDim.x + threadIdx.x;
    if (id >= EN * H) return;
    const int e = id >> 2, hh = id & 3;
    int src, dst;
    edge_endpoints(ei, e, src, dst);
    float v = a_s[src * H + hh] + a_d[dst * H + hh];
    v = (v > 0.0f) ? v : SLOPE * v;
    ebuf[id] = v;
    atomicMaxF(&emax[dst * H + hh], v);
}

// pass 2: p = exp(e - emax[dst]); segment sum -> denom
__global__ void edge_pass2(const int* __restrict__ ei, float* __restrict__ ebuf,
                           const float* __restrict__ emax, float* __restrict__ denom) {
    const int id = blockIdx.x * blockDim.x + threadIdx.x;
    if (id >= EN * H) return;
    const int e = id >> 2, hh = id & 3;
    int src, dst;
    edge_endpoints(ei, e, src, dst);
    const float p = expf(ebuf[id] - emax[dst * H + hh]);
    ebuf[id] = p;
    atomicAdd(&denom[dst * H + hh], p);
}

// pass 3: agg[dst] += (p/denom[dst]) * h[src]   (block per edge, 512 channels)
__global__ __launch_bounds__(256)
void edge_pass3(const int* __restrict__ ei, const float* __restrict__ ebuf,
                const float* __restrict__ denom, const float* __restrict__ h,
                float* __restrict__ agg) {
    const int e = blockIdx.x;
    int src, dst;
    edge_endpoints(ei, e, src, dst);
    const float* hs = h + (size_t)src * HC;
    float* ag = agg + (size_t)dst * HC;
    for (int cc = threadIdx.x; cc < HC; cc += 256) {
        const int hh = cc >> 7;
        const float alpha = ebuf[e * H + hh] / denom[dst * H + hh];
        atomicAdd(&ag[cc], alpha * hs[cc]);
    }
}

// g = mean over heads of agg + bias
__global__ void head_mean_kernel(const float* __restrict__ agg,
                                 const float* __restrict__ bias,
                                 float* __restrict__ g) {
    const int id = blockIdx.x * blockDim.x + threadIdx.x;
    if (id >= N_NODES * C) return;
    const int i = id >> 7;
    const int c = id & (C - 1);
    const float* a = agg + (size_t)i * HC + c;
    g[id] = 0.25f * (a[0] + a[C] + a[2 * C] + a[3 * C]) + bias[c];
}

// ---------------------------------------------------------------------------
extern "C" void kernel_launch(void* const* d_in, const int* in_sizes, int n_in,
                              void* d_out, int out_size, void* d_ws, size_t ws_size,
                              hipStream_t stream) {
    const float* x       = (const float*)d_in[0];
    const int*   ei      = (const int*)  d_in[1];
    const float* W       = (const float*)d_in[2];
    const float* att_src = (const float*)d_in[3];
    const float* att_dst = (const float*)d_in[4];
    const float* bias    = (const float*)d_in[5];
    const float* Wr      = (const float*)d_in[6];
    const float* br      = (const float*)d_in[7];
    const float* w0 = (const float*)d_in[8],  *b0 = (const float*)d_in[9];
    const float* w1 = (const float*)d_in[10], *b1 = (const float*)d_in[11];
    const float* w2 = (const float*)d_in[12], *b2 = (const float*)d_in[13];
    const float* w3 = (const float*)d_in[14], *b3 = (const float*)d_in[15];

    float* out = (float*)d_out;
    float* ws  = (float*)d_ws;

    size_t o = 0;
    float* h_buf = ws + o;  o += (size_t)N_NODES * HC;   // [N,512]
    float* a_s   = ws + o;  o += (size_t)N_NODES * H;
    float* a_d   = ws + o;  o += (size_t)N_NODES * H;
    float* ebuf  = ws + o;  o += (size_t)EN * H;
    float* emax  = ws + o;  o += (size_t)N_NODES * H;
    float* denom = ws + o;  o += (size_t)N_NODES * H;
    float* agg   = ws + o;  o += (size_t)N_NODES * HC;   // [N,512]
    float* gbuf  = ws + o;  o += (size_t)N_NODES * C;    // [N,128]

    const int MT = N_NODES / 16;                 // 3125 M tiles
    const int nh_blocks = (N_NODES * H + 255) / 256;
    const int eh_blocks = (EN * H + 255) / 256;
    const int nc_blocks = (N_NODES * C + 255) / 256;

    // 0) clear reduction buffers
    init_kernel<<<nh_blocks, 256, 0, stream>>>(emax, denom, agg);

    // 1) h = x @ W^T   [N,512], fp32 WMMA, TN=4 (wave strip 16x64)
    gemm_wmma<0, 4><<<dim3(MT, 2), 128, 0, stream>>>(
        x, W, nullptr, nullptr, h_buf, 128, HC);

    // 2) per-(node,head) scores
    att_score_kernel<<<nh_blocks, 256, 0, stream>>>(h_buf, att_src, att_dst, a_s, a_d);

    // 3) segment softmax over incoming edges
    edge_pass1<<<eh_blocks, 256, 0, stream>>>(ei, a_s, a_d, ebuf, emax);
    edge_pass2<<<eh_blocks, 256, 0, stream>>>(ei, ebuf, emax, denom);
    edge_pass3<<<EN, 256, 0, stream>>>(ei, ebuf, denom, h_buf, agg);

    // 4) g = head-mean(agg) + bias
    head_mean_kernel<<<nc_blocks, 256, 0, stream>>>(agg, bias, gbuf);

    // 5) xr = relu(g @ Wr^T + br + g)  -> out slice 0   (TN=2, Nout=128)
    float* xr = out;
    gemm_wmma<1, 2><<<dim3(MT, 1), 128, 0, stream>>>(
        gbuf, Wr, br, gbuf, xr, C, C);

    // 6) tanh MLP stack, each activation into its out slice
    float* act1 = out + (size_t)N_NODES * 128;
    float* act2 = out + (size_t)N_NODES * 256;
    float* act3 = out + (size_t)N_NODES * 768;
    float* act4 = out + (size_t)N_NODES * 1280;

    gemm_wmma<2, 2><<<dim3(MT, 1), 128, 0, stream>>>(
        xr,   w0, b0, nullptr, act1, 128, 128);
    gemm_wmma<2, 4><<<dim3(MT, 2), 128, 0, stream>>>(
        act1, w1, b1, nullptr, act2, 128, 512);
    gemm_wmma<2, 4><<<dim3(MT, 2), 128, 0, stream>>>(
        act2, w2, b2, nullptr, act3, 512, 512);
    gemm_wmma<2, 2><<<dim3(MT, 1), 128, 0, stream>>>(
        act3, w3, b3, nullptr, act4, 512, 128);
}